// KMeansBase_15771119911642
// MI455X (gfx1250) — compile-verified
//
#include <hip/hip_runtime.h>
#include <hip/hip_bf16.h>
#include <cstdint>

#define BB    4
#define N_PTS 32768
#define FDIM  64
#define KC    8
#define ROWS  128          // rows per block in assign kernel (8 waves x 16)
#define XP    68           // padded LDS row pitch for X tile (floats)
#define CP    68           // padded LDS row pitch for centroid tile (floats)
#define GSTEP 64           // blocks per batch in step kernels (64*256 == N/2 pairs)

typedef float v2f __attribute__((ext_vector_type(2)));
typedef float v8f __attribute__((ext_vector_type(8)));

struct RNGArgs {
  unsigned sk0[BB][7];   // step keys (Gumbel) per batch, steps 1..7
  unsigned sk1[BB][7];
  int first[BB];         // data-independent first centroid id per batch
};

// ---------------- Threefry2x32 (exactly JAX's PRF), host + device ----------------
__host__ __device__ __forceinline__ void tf2x32(unsigned k0, unsigned k1,
                                                unsigned x0, unsigned x1,
                                                unsigned &o0, unsigned &o1) {
  unsigned ks2 = k0 ^ k1 ^ 0x1BD11BDAu;
  x0 += k0; x1 += k1;
#define RND(r) { x0 += x1; x1 = (x1 << (r)) | (x1 >> (32 - (r))); x1 ^= x0; }
#define R4A RND(13) RND(15) RND(26) RND(6)
#define R4B RND(17) RND(29) RND(16) RND(24)
  R4A x0 += k1;  x1 += ks2 + 1u;
  R4B x0 += ks2; x1 += k0 + 2u;
  R4A x0 += k0;  x1 += k1 + 3u;
  R4B x0 += k1;  x1 += ks2 + 4u;
  R4A x0 += ks2; x1 += k0 + 5u;
#undef R4A
#undef R4B
#undef RND
  o0 = x0; o1 = x1;
}

// bits -> uniform[tiny,1) -> standard Gumbel, matching jax.random.gumbel
__device__ __forceinline__ float gumbel_from_bits(unsigned bits) {
  float u = __uint_as_float((bits >> 9) | 0x3F800000u) - 1.0f;   // [0,1)
  const float tiny = 1.175494350822287508e-38f;
  float f = fmaxf(tiny, u + tiny);    // u*(1-tiny)+tiny ; (1-tiny) rounds to 1
  return -logf(-logf(f));
}

// monotonic packing: larger score wins; equal scores -> smaller index wins
__device__ __forceinline__ unsigned long long pack_score(float v, int n) {
  unsigned u = __float_as_uint(v);
  u = (u & 0x80000000u) ? ~u : (u | 0x80000000u);
  return ((unsigned long long)u << 32) | (unsigned)(N_PTS - 1 - n);
}

__device__ __forceinline__ float dist_to(const float* __restrict__ x, int n,
                                         const float* __restrict__ c) {
  const float4* xr = (const float4*)(x + (size_t)n * FDIM);
  float acc = 0.0f;
#pragma unroll
  for (int j = 0; j < FDIM / 4; ++j) {
    float4 v = xr[j];
    float d0 = v.x - c[4 * j + 0];
    float d1 = v.y - c[4 * j + 1];
    float d2 = v.z - c[4 * j + 2];
    float d3 = v.w - c[4 * j + 3];
    acc = fmaf(d0, d0, acc);
    acc = fmaf(d1, d1, acc);
    acc = fmaf(d2, d2, acc);
    acc = fmaf(d3, d3, acc);
  }
  return sqrtf(acc);
}

// ---------------- zero the argmax slots (must be re-done every call) ----------------
__global__ void kpp_init(unsigned long long* __restrict__ slots) {
  const int t = threadIdx.x;
  if (t < BB * KC) slots[t] = 0ull;
}

// ---------------- one kmeans++ step, full-GPU parallel ----------------
// step = 1..7.  Updates min_d with centroid (step-1), scores log(min_d)+gumbel,
// argmax across the grid via atomicMax on the monotonic packing.
__global__ __launch_bounds__(256)
void kpp_step(const float* __restrict__ data, float* __restrict__ mind,
              unsigned long long* __restrict__ slots, RNGArgs args, int step) {
  __shared__ __align__(16) float sc[FDIM];
  __shared__ unsigned long long rp[256];

  const int b = blockIdx.x / GSTEP;
  const int g = blockIdx.x % GSTEP;
  const int t = threadIdx.x;
  const float* x = data + (size_t)b * N_PTS * FDIM;
  float* md = mind + (size_t)b * N_PTS;

  int nid;
  if (step == 1) {
    nid = args.first[b];
  } else {
    const unsigned long long pk = slots[b * KC + (step - 1)];
    nid = N_PTS - 1 - (int)(unsigned)(pk & 0xFFFFFFFFull);
  }
  if (t < FDIM) sc[t] = x[(size_t)nid * FDIM + t];
  __syncthreads();

  const int p  = g * 256 + t;          // pair index 0..16383
  const int n0 = p, n1 = p + N_PTS / 2;
  float d0 = dist_to(x, n0, sc);
  float d1 = dist_to(x, n1, sc);
  if (step > 1) { d0 = fminf(d0, md[n0]); d1 = fminf(d1, md[n1]); }
  md[n0] = d0; md[n1] = d1;

  // threefry counter pair (p, p+N/2) yields bits for points n0 and n1
  unsigned g0, g1;
  tf2x32(args.sk0[b][step - 1], args.sk1[b][step - 1],
         (unsigned)p, (unsigned)n1, g0, g1);
  const float v0 = logf(d0) + gumbel_from_bits(g0);
  const float v1 = logf(d1) + gumbel_from_bits(g1);
  const unsigned long long pk0 = pack_score(v0, n0);
  const unsigned long long pk1 = pack_score(v1, n1);
  rp[t] = pk0 > pk1 ? pk0 : pk1;
  __syncthreads();
  for (int s = 128; s > 0; s >>= 1) {
    if (t < s) { if (rp[t + s] > rp[t]) rp[t] = rp[t + s]; }
    __syncthreads();
  }
  if (t == 0) atomicMax(&slots[b * KC + step], rp[0]);
}

// ---------------- assignment: WMMA f32 16x16x4, async global->LDS staging ----------------
__global__ __launch_bounds__(256)
void kmeans_assign(const float* __restrict__ data,
                   const unsigned long long* __restrict__ slots,
                   RNGArgs args, int* __restrict__ out) {
  __shared__ __align__(16) float sxt[ROWS * XP];   // 128x64 X tile, padded rows
  __shared__ __align__(16) float scp[16 * CP];     // centroids^T, 16 cols x 64 k
  __shared__ float scn[16];
  __shared__ int sids[KC];

  const int b    = blockIdx.x / (N_PTS / ROWS);
  const int tile = blockIdx.x % (N_PTS / ROWS);
  const int t = threadIdx.x;
  const float* xb = data + ((size_t)b * N_PTS + (size_t)tile * ROWS) * FDIM;

  // async global->LDS copy of the 128x64 tile (CDNA5 path, ASYNCcnt-tracked),
  // per-lane LDS destinations preserve the bank-conflict-avoiding row pad
#pragma unroll
  for (int j = 0; j < 8; ++j) {
    const int q = t + 256 * j;               // float4 index 0..2047
    const int row = q >> 4;
    const int col = (q & 15) << 2;
    const unsigned lds = (unsigned)(uintptr_t)(&sxt[row * XP + col]);
    const unsigned voff = (unsigned)(q * 16);
    asm volatile("global_load_async_to_lds_b128 %0, %1, %2"
                 :
                 : "v"(lds), "v"(voff),
                   "s"((unsigned long long)(uintptr_t)xb)
                 : "memory");
  }

  // decode the 8 centroid ids (k=0 is data-independent, k=1..7 from slots)
  if (t < KC)
    sids[t] = (t == 0) ? args.first[b]
                       : (N_PTS - 1 - (int)(unsigned)(slots[b * KC + t] & 0xFFFFFFFFull));
  __syncthreads();

  // centroid tile, transposed: scp[col][k]; cols 8..15 zero-padded
#pragma unroll
  for (int j = 0; j < 4; ++j) {
    const int q = t + 256 * j;               // 0..1023
    const int col = q >> 6;
    const int k = q & 63;
    scp[col * CP + k] = (col < KC)
        ? data[((size_t)b * N_PTS + sids[col]) * FDIM + k] : 0.0f;
  }
  asm volatile("s_wait_asynccnt 0x0" ::: "memory");
  __syncthreads();

  if (t < 16) {                              // ||c_col||^2
    float s = 0.0f;
#pragma unroll
    for (int k = 0; k < FDIM; ++k) { const float v = scp[t * CP + k]; s = fmaf(v, v, s); }
    scn[t] = s;
  }
  __syncthreads();

  const int w = t >> 5;                  // wave 0..7 -> rows 16w..16w+15
  const int l = t & 31;
  const int m = l & 15;
  const int koff = (l < 16) ? 0 : 2;     // fp32 A/B fragment half-split over K
  const float* arow = &sxt[(w * 16 + m) * XP];
  const float* brow = &scp[m * CP];

  v8f acc = {};
#pragma unroll
  for (int tk = 0; tk < 16; ++tk) {      // K = 64 = 16 * 4
    const v2f a  = *(const v2f*)(arow + 4 * tk + koff);
    const v2f bb = *(const v2f*)(brow + 4 * tk + koff);
    acc = __builtin_amdgcn_wmma_f32_16x16x4_f32(
        false, a, false, bb, (short)0, acc, false, false);
  }

  // score for column m: -2*dot + ||c_m||^2 ; padded cols -> +inf
  const bool valid = (m < KC);
  const float cn = valid ? scn[m] : 0.0f;
  const int base = tile * ROWS + w * 16;

  int bestidx[8];
#pragma unroll
  for (int r = 0; r < 8; ++r) {
    float val = valid ? fmaf(-2.0f, acc[r], cn) : __builtin_inff();
    int idx = m;
#pragma unroll
    for (int mask = 1; mask < 16; mask <<= 1) {      // stays within 16-lane half
      const float ov = __shfl_xor(val, mask, 32);
      const int   oi = __shfl_xor(idx, mask, 32);
      if (ov < val || (ov == val && oi < idx)) { val = ov; idx = oi; }
    }
    bestidx[r] = idx;
  }
  if (m == 0) {                       // lane 0 -> rows M=0..7, lane 16 -> M=8..15
    const int roff = (l < 16) ? 0 : 8;
    const size_t ob = (size_t)b * N_PTS + base + roff;
#pragma unroll
    for (int r = 0; r < 8; ++r) out[ob + r] = bestidx[r];
  }
}

// ---------------- host-side RNG: keys + first ids are data-independent ----------------
static void fill_rng(RNGArgs& a) {
  // root = key(42) = (0,42); split(root, 4): counts iota(8) -> pairs (j, j+4)
  unsigned o0[4], o1[4];
  for (int j = 0; j < 4; ++j) tf2x32(0u, 42u, (unsigned)j, (unsigned)(j + 4), o0[j], o1[j]);
  const unsigned bk[BB][2] = { {o0[0], o0[1]}, {o0[2], o0[3]}, {o1[0], o1[1]}, {o1[2], o1[3]} };
  for (int b = 0; b < BB; ++b) {
    // k0, k1 = split(bkey): counts iota(4) -> pairs (0,2),(1,3)
    unsigned e0, e1, f0, f1;
    tf2x32(bk[b][0], bk[b][1], 0u, 2u, e0, e1);
    tf2x32(bk[b][0], bk[b][1], 1u, 3u, f0, f1);
    // first = randint(k0, (), 0, 2^15): multiplier==0 -> low16 % 2^15
    unsigned rb0, rb1;
    tf2x32(e0, f0, 0u, 0u, rb0, rb1);
    a.first[b] = (int)(rb0 & 0x7FFFu);
    // step_keys = split(k1, 7): counts iota(14) -> pairs (j, j+7), reshape(7,2)
    unsigned h0[7], h1[7];
    for (int j = 0; j < 7; ++j) tf2x32(e1, f1, (unsigned)j, (unsigned)(j + 7), h0[j], h1[j]);
    const unsigned flat[14] = { h0[0], h0[1], h0[2], h0[3], h0[4], h0[5], h0[6],
                                h1[0], h1[1], h1[2], h1[3], h1[4], h1[5], h1[6] };
    for (int i = 0; i < 7; ++i) { a.sk0[b][i] = flat[2 * i]; a.sk1[b][i] = flat[2 * i + 1]; }
  }
}

extern "C" void kernel_launch(void* const* d_in, const int* in_sizes, int n_in,
                              void* d_out, int out_size, void* d_ws, size_t ws_size,
                              hipStream_t stream) {
  const float* data = (const float*)d_in[0];
  int* out = (int*)d_out;

  char* ws = (char*)d_ws;
  float* mind = (float*)ws;                                             // B*N f32
  unsigned long long* slots = (unsigned long long*)(ws + (size_t)BB * N_PTS * 4);

  RNGArgs args;
  fill_rng(args);

  kpp_init<<<1, 64, 0, stream>>>(slots);
  for (int step = 1; step < KC; ++step)
    kpp_step<<<BB * GSTEP, 256, 0, stream>>>(data, mind, slots, args, step);
  kmeans_assign<<<BB * (N_PTS / ROWS), 256, 0, stream>>>(data, slots, args, out);
}